// GATStage4_AttentionSoftmax_ONNX_51994874085821
// MI455X (gfx1250) — compile-verified
//
#include <hip/hip_runtime.h>
#include <hip/hip_bf16.h>

// Segment softmax of edge scores over destination nodes.
// E = 32768 edges, N = 4096 nodes. Entirely L2-resident on MI455X (192 MB L2);
// O(E) scatter-atomic formulation instead of the reference's O(E*N) one-hot
// matmul emulation. Latency-bound: ~0.9 MB total traffic (~40 ns at 23.3 TB/s).

static constexpr int NUM_NODES = 4096;

// Order-preserving float->u32 mapping so scatter-max uses a single native
// GLOBAL_ATOMIC_MAX_U32 (no CAS loop). Monotonic: a < b  <=>  flip(a) < flip(b).
__device__ __forceinline__ unsigned flipF32(float f) {
  unsigned u = __float_as_uint(f);
  return (u & 0x80000000u) ? ~u : (u | 0x80000000u);
}
__device__ __forceinline__ float unflipF32(unsigned k) {
  unsigned u = (k & 0x80000000u) ? (k ^ 0x80000000u) : ~k;
  return __uint_as_float(u);
}

// ---------------------------------------------------------------------------
// Pass 0: init per-node accumulators in workspace.
// maxK identity = 0u (smallest flipped key); sum identity = 0.0f.
__global__ __launch_bounds__(256) void k_init_nodes(unsigned* __restrict__ maxK,
                                                    float* __restrict__ sumE) {
  int i = blockIdx.x * blockDim.x + threadIdx.x;
  if (i < NUM_NODES) {
    maxK[i] = 0u;
    sumE[i] = 0.0f;
  }
}

// ---------------------------------------------------------------------------
// Pass 1: scatter-max of edge scores into per-node max.
// Edge slice is staged via CDNA5 async global->LDS loads (ASYNCcnt path),
// then each lane reads back its own LDS slot after s_wait_asynccnt 0.
__global__ __launch_bounds__(256) void k_scatter_max(const float* __restrict__ e,
                                                     const int* __restrict__ dst,
                                                     unsigned* __restrict__ maxK,
                                                     int E) {
  __shared__ float sh_e[256];
  __shared__ int   sh_d[256];
  int i = blockIdx.x * blockDim.x + threadIdx.x;
  if (i < E) {
    // Low 32 bits of a generic pointer to __shared__ == wave-relative LDS
    // byte offset (flat->LDS aperture mapping uses addr[31:0]).
    unsigned ldsE = (unsigned)(size_t)(&sh_e[threadIdx.x]);
    unsigned ldsD = (unsigned)(size_t)(&sh_d[threadIdx.x]);
    unsigned long long gaE = (unsigned long long)(size_t)(e + i);
    unsigned long long gaD = (unsigned long long)(size_t)(dst + i);
    asm volatile("global_load_async_to_lds_b32 %0, %1, off"
                 :: "v"(ldsE), "v"(gaE) : "memory");
    asm volatile("global_load_async_to_lds_b32 %0, %1, off"
                 :: "v"(ldsD), "v"(gaD) : "memory");
    asm volatile("s_wait_asynccnt 0" ::: "memory");
    float v = sh_e[threadIdx.x];
    int   t = sh_d[threadIdx.x];
    atomicMax(&maxK[t], flipF32(v));   // -> global_atomic_max_u32
  }
}

// ---------------------------------------------------------------------------
// Pass 2: exp(e - max[dst]) scatter-added into per-node sum.
__global__ __launch_bounds__(256) void k_scatter_sum(const float* __restrict__ e,
                                                     const int* __restrict__ dst,
                                                     const unsigned* __restrict__ maxK,
                                                     float* __restrict__ sumE,
                                                     int E) {
  int i = blockIdx.x * blockDim.x + threadIdx.x;
  if (i < E) {
    int   t  = dst[i];
    float m  = unflipF32(maxK[t]);
    float ex = expf(e[i] - m);          // arg <= 0, no overflow
    atomicAdd(&sumE[t], ex);            // -> global_atomic_add_f32
  }
}

// ---------------------------------------------------------------------------
// Pass 3: alpha = exp(e - max[dst]) / (sum[dst] + 1e-16).
// Recomputing expf (one v_exp_f32) beats a 128 KB scratch round trip.
__global__ __launch_bounds__(256) void k_normalize(const float* __restrict__ e,
                                                   const int* __restrict__ dst,
                                                   const unsigned* __restrict__ maxK,
                                                   const float* __restrict__ sumE,
                                                   float* __restrict__ out,
                                                   int E) {
  int i = blockIdx.x * blockDim.x + threadIdx.x;
  if (i < E) {
    int   t  = dst[i];
    float m  = unflipF32(maxK[t]);
    float ex = expf(e[i] - m);
    out[i]   = ex / (sumE[t] + 1e-16f);
  }
}

// ---------------------------------------------------------------------------
extern "C" void kernel_launch(void* const* d_in, const int* in_sizes, int n_in,
                              void* d_out, int out_size, void* d_ws, size_t ws_size,
                              hipStream_t stream) {
  const float* e          = (const float*)d_in[0];       // [E]
  const int*   edge_index = (const int*)d_in[1];         // [2, E] row-major
  const int    E          = in_sizes[0];
  const int*   dst        = edge_index + E;              // row 1 = destinations

  // Workspace layout: [0, N) flipped-u32 per-node max, [N, 2N) f32 per-node sum.
  unsigned* maxK = (unsigned*)d_ws;
  float*    sumE = (float*)d_ws + NUM_NODES;

  float* out = (float*)d_out;

  const int BLK    = 256;                  // 8 wave32 waves per workgroup
  const int gNodes = (NUM_NODES + BLK - 1) / BLK;
  const int gEdges = (E + BLK - 1) / BLK;

  k_init_nodes <<<gNodes, BLK, 0, stream>>>(maxK, sumE);
  k_scatter_max<<<gEdges, BLK, 0, stream>>>(e, dst, maxK, E);
  k_scatter_sum<<<gEdges, BLK, 0, stream>>>(e, dst, maxK, sumE, E);
  k_normalize  <<<gEdges, BLK, 0, stream>>>(e, dst, maxK, sumE, out, E);
}